// PINN_GAN_70712341561989
// MI455X (gfx1250) — compile-verified
//
#include <hip/hip_runtime.h>
#include <hip/hip_bf16.h>
#include <math.h>

// CDNA5 (gfx1250) fused PINN forward + nested-JVP kernel.
// Three value streams (h, dh/dx0, d2h/dx0^2) are pushed through the MLP with
// v_wmma_f32_16x16x4_f32; activations stay in LDS, accumulators in registers.

typedef __attribute__((ext_vector_type(2))) float v2f;
typedef __attribute__((ext_vector_type(8))) float v8f;

#define NPTS   262144
#define HID    128
#define MTILE  16
#define LSTR   132   // padded LDS row stride in floats (132 % 64 = 4 -> bank spread)

__global__ __launch_bounds__(256) void pinn_fused_wmma(
    const float* __restrict__ x,
    const float* __restrict__ W0, const float* __restrict__ b0,
    const float* __restrict__ W1, const float* __restrict__ b1,
    const float* __restrict__ W2, const float* __restrict__ b2,
    const float* __restrict__ W3, const float* __restrict__ b3,
    const float* __restrict__ W4, const float* __restrict__ b4,
    float* __restrict__ out)
{
    __shared__ float hbuf[6][MTILE * LSTR];   // {h,h1,h2} x {cur,nxt}
    __shared__ float res[3][MTILE][2];        // final-layer partials

    const int tid  = threadIdx.x;
    const int wave = tid >> 5;
    const int lane = tid & 31;
    const int p0   = blockIdx.x * MTILE;

    // ---------------- layer 0: 2 -> 128, tanh + derivative seeding -------------
    // z0 = (x0+5)/10*2-1 = 0.2*x0 ; z1 = x1*(2/1.57)-1 ; dz0/dx0 = 0.2, dz1/dx0 = 0
    const float s0 = 0.2f;
    for (int e = tid; e < MTILE * HID; e += 256) {
        const int m = e >> 7, n = e & 127;
        const float x0 = x[(p0 + m) * 2 + 0];
        const float x1 = x[(p0 + m) * 2 + 1];
        const float z0 = 0.2f * x0;
        const float z1 = x1 * (2.0f / 1.57f) - 1.0f;
        const float w0n = W0[n], w1n = W0[HID + n];
        const float a  = z0 * w0n + z1 * w1n + b0[n];
        const float a1 = s0 * w0n;                 // tangent, a2 = 0
        const float t  = tanhf(a);
        const float g  = 1.0f - t * t;
        hbuf[0][m * LSTR + n] = t;
        hbuf[1][m * LSTR + n] = g * a1;
        hbuf[2][m * LSTR + n] = -2.0f * t * g * a1 * a1;  // g*(0 - 2*t*a1^2)
    }
    __syncthreads();

    const float* Ws[3] = { W1, W2, W3 };
    const float* bs[3] = { b1, b2, b3 };

    int cur = 0, nxt = 3;
    const int n0   = wave * 16;          // this wave's 16-wide feature slice
    const int hrow = lane & 15;          // A: M row ; B/C/D: N column
    const int kk   = (lane >> 4) * 2;    // K sub-offset per lane-half (ISA 7.12.2)
    const int mhi  = (lane >> 4) * 8;    // C/D: M offset per lane-half

    // ---------------- layers 1..3: 128 -> 128 via WMMA f32 ---------------------
    for (int l = 0; l < 3; ++l) {
        const float* __restrict__ W = Ws[l];
        const float* __restrict__ b = bs[l];
        if (l < 2) __builtin_prefetch(Ws[l + 1] + (tid << 6), 0, 1); // warm L2 for next layer

        v8f acc_h  = {};
        v8f acc_d1 = {};
        v8f acc_d2 = {};

        const float* ha  = &hbuf[cur + 0][hrow * LSTR];
        const float* d1a = &hbuf[cur + 1][hrow * LSTR];
        const float* d2a = &hbuf[cur + 2][hrow * LSTR];

        #pragma unroll 8
        for (int k = 0; k < HID; k += 4) {
            const int kz = k + kk;
            // B operand (4x16 f32): rows K=kz,kz+1 of W, column n0+hrow. Shared
            // by all three streams (weights are identical per stream).
            v2f bv;
            bv.x = W[(kz    ) * HID + n0 + hrow];
            bv.y = W[(kz + 1) * HID + n0 + hrow];

            v2f av;
            av.x = ha[kz]; av.y = ha[kz + 1];
            acc_h  = __builtin_amdgcn_wmma_f32_16x16x4_f32(
                         false, av, false, bv, (short)0, acc_h,  false, false);
            av.x = d1a[kz]; av.y = d1a[kz + 1];
            acc_d1 = __builtin_amdgcn_wmma_f32_16x16x4_f32(
                         false, av, false, bv, (short)0, acc_d1, false, false);
            av.x = d2a[kz]; av.y = d2a[kz + 1];
            acc_d2 = __builtin_amdgcn_wmma_f32_16x16x4_f32(
                         false, av, false, bv, (short)0, acc_d2, false, false);
        }

        // Fused bias + tanh + 1st/2nd-derivative chain rule on the accumulators.
        const float bias = b[n0 + hrow];
        #pragma unroll
        for (int r = 0; r < 8; ++r) {
            const int m   = r + mhi;
            const int off = m * LSTR + n0 + hrow;
            const float a  = acc_h[r] + bias;
            const float d1 = acc_d1[r];
            const float d2 = acc_d2[r];
            const float t  = tanhf(a);
            const float g  = 1.0f - t * t;
            hbuf[nxt + 0][off] = t;
            hbuf[nxt + 1][off] = g * d1;
            hbuf[nxt + 2][off] = g * (d2 - 2.0f * t * d1 * d1);
        }
        __syncthreads();
        const int tsw = cur; cur = nxt; nxt = tsw;
    }

    // ---------------- layer 4: 128 -> 2 (linear), all 3 streams ----------------
    if (tid < 96) {
        const int m  = tid & 15;
        const int pr = tid >> 4;     // 0..5
        const int j  = pr & 1;       // output column
        const int st = pr >> 1;      // stream 0..2
        const float* hv = &hbuf[cur + st][m * LSTR];
        float acc = 0.0f;
        #pragma unroll 8
        for (int k = 0; k < HID; ++k) acc += hv[k] * W4[k * 2 + j];
        if (st == 0) acc += b4[j];
        res[st][m][j] = acc;
    }
    __syncthreads();

    // ---------------- residual assembly + store --------------------------------
    if (tid < MTILE) {
        const int m = tid, p = p0 + m;
        const float u   = res[0][m][0], v   = res[0][m][1];
        const float du  = res[1][m][0], dv  = res[1][m][1];
        const float d2u = res[2][m][0], d2v = res[2][m][1];
        const float mag = u * u + v * v;
        out[p * 2 + 0] = u;
        out[p * 2 + 1] = v;
        out[2 * NPTS + p * 2 + 0] = du + 0.5f * d2u + mag * v;
        out[2 * NPTS + p * 2 + 1] = dv - 0.5f * d2v - mag * u;
    }
}

extern "C" void kernel_launch(void* const* d_in, const int* in_sizes, int n_in,
                              void* d_out, int out_size, void* d_ws, size_t ws_size,
                              hipStream_t stream) {
    (void)in_sizes; (void)n_in; (void)out_size; (void)d_ws; (void)ws_size;
    const float* x  = (const float*)d_in[0];
    const float* W0 = (const float*)d_in[1];
    const float* b0 = (const float*)d_in[2];
    const float* W1 = (const float*)d_in[3];
    const float* b1 = (const float*)d_in[4];
    const float* W2 = (const float*)d_in[5];
    const float* b2 = (const float*)d_in[6];
    const float* W3 = (const float*)d_in[7];
    const float* b3 = (const float*)d_in[8];
    const float* W4 = (const float*)d_in[9];
    const float* b4 = (const float*)d_in[10];

    pinn_fused_wmma<<<NPTS / MTILE, 256, 0, stream>>>(
        x, W0, b0, W1, b1, W2, b2, W3, b3, W4, b4, (float*)d_out);
}